// CayleyLinear_69630009803033
// MI455X (gfx1250) — compile-verified
//
#include <hip/hip_runtime.h>

typedef __attribute__((ext_vector_type(2))) float v2f;
typedef __attribute__((ext_vector_type(8))) float v8f;

#define DIM 2048
#define BATCH_ROWS 8192   // 4 * 2048
#define NEWTON_ITERS 26

#define TM 128
#define TN 128
#define TK 32
#define LDSA 36           // TK + 4 pad: 16B-aligned b128 async stores, conflict-free b64 frag loads
#define LDSB_NN 132       // TN + 4 pad

// ---------------------------------------------------------------------------
// CDNA5 async global->LDS copy (ASYNCcnt-tracked, bypasses VGPRs).
// LDS operand is the 32-bit LDS byte offset (low 32 bits of a generic shared
// pointer, per the gfx1250 aperture mapping).
// ---------------------------------------------------------------------------
__device__ __forceinline__ void async_g2l_b128(const float* lds_generic, const float* gptr)
{
    const unsigned lds_addr = (unsigned)(size_t)lds_generic;
    asm volatile("global_load_async_to_lds_b128 %0, %1, off"
                 :: "v"(lds_addr), "v"(gptr)
                 : "memory");
}

__device__ __forceinline__ void wait_async_all()
{
    asm volatile("s_wait_asynccnt 0x0" ::: "memory");
}

// ---------------------------------------------------------------------------
// f32 WMMA GEMM:  D = epilogue( A(MxK) * B' ) where
//   TRANS_B = false : B is KxN row-major          (D = A*B)
//   TRANS_B = true  : B is NxK row-major, use B^T (D = A*B^T)
// EPI = 0 : D = acc
// EPI = 1 : D = alpha*acc + beta*C      (C = aux, ldc = Ndim)
// EPI = 2 : D = acc + bias[col]         (bias = aux)
// Requires Mdim%128==0, Ndim%128==0, Kdim%32==0 (true for all uses here).
// Double-buffered LDS; tile kt+1 is DMA'd (async-to-LDS) under tile kt's WMMAs.
// ---------------------------------------------------------------------------
template<bool TRANS_B, int EPI>
__global__ __launch_bounds__(256)
void gemm_f32_wmma(const float* __restrict__ A, const float* __restrict__ B,
                   float* __restrict__ D, const float* __restrict__ aux,
                   float alpha, float beta,
                   int Mdim, int Ndim, int Kdim)
{
    constexpr int SB_ELEMS = TRANS_B ? (TN * LDSA) : (TK * LDSB_NN);
    __shared__ float sA[2][TM * LDSA];
    __shared__ float sB[2][SB_ELEMS];

    const int tid   = threadIdx.x;
    const int lane  = tid & 31;
    const int wave  = tid >> 5;
    const int waveM = (wave & 1) * 64;    // 2 waves along M
    const int waveN = (wave >> 1) * 32;   // 4 waves along N
    const int ln    = lane & 15;
    const int khalf = lane >> 4;          // 0: K pair {k,k+1}; 1: {k+2,k+3}

    const int bm = blockIdx.x * TM;
    const int bn = blockIdx.y * TN;

    // Issue the 8 async b128 copies (A: 128x32, B: 128x32 or 32x128) for tile kt
    // into LDS buffer `buf`. 256 threads x 4 slots cover each 1024-float4 tile.
    auto issueTile = [&](int kt, int buf) {
        const float* gA = A + (size_t)bm * Kdim + kt * TK;
#pragma unroll
        for (int it = 0; it < 4; ++it) {
            const int idx = tid + it * 256;
            const int row = idx >> 3;          // 8 float4 per 32-wide row
            const int c4  = (idx & 7) << 2;
            async_g2l_b128(&sA[buf][row * LDSA + c4], gA + (size_t)row * Kdim + c4);
        }
        if (TRANS_B) {
            const float* gB = B + (size_t)bn * Kdim + kt * TK;
#pragma unroll
            for (int it = 0; it < 4; ++it) {
                const int idx = tid + it * 256;
                const int row = idx >> 3;
                const int c4  = (idx & 7) << 2;
                async_g2l_b128(&sB[buf][row * LDSA + c4], gB + (size_t)row * Kdim + c4);
            }
        } else {
            const float* gB = B + (size_t)(kt * TK) * Ndim + bn;
#pragma unroll
            for (int it = 0; it < 4; ++it) {
                const int idx = tid + it * 256;
                const int row = idx >> 5;      // 32 float4 per 128-wide row
                const int c4  = (idx & 31) << 2;
                async_g2l_b128(&sB[buf][row * LDSB_NN + c4], gB + (size_t)row * Ndim + c4);
            }
        }
    };

    v8f acc[4][2];
    const v8f zero8 = {0.f,0.f,0.f,0.f,0.f,0.f,0.f,0.f};
#pragma unroll
    for (int mi = 0; mi < 4; ++mi)
#pragma unroll
        for (int ni = 0; ni < 2; ++ni)
            acc[mi][ni] = zero8;

    // ---- pipeline prologue: land tile 0 ----
    issueTile(0, 0);
    wait_async_all();
    __syncthreads();

    const int ktiles = Kdim / TK;
    for (int kt = 0; kt < ktiles; ++kt) {
        const int cur = kt & 1;
        if (kt + 1 < ktiles)
            issueTile(kt + 1, cur ^ 1);   // DMA next tile under this tile's WMMAs

        // ---- 8 K-steps of V_WMMA_F32_16X16X4_F32 on buffer `cur` ----
#pragma unroll
        for (int ks = 0; ks < TK / 4; ++ks) {
            const int kk = ks * 4 + khalf * 2;
            v2f aF[4], bF[2];
#pragma unroll
            for (int mi = 0; mi < 4; ++mi)
                aF[mi] = *(const v2f*)(&sA[cur][(waveM + mi * 16 + ln) * LDSA + kk]);
#pragma unroll
            for (int ni = 0; ni < 2; ++ni) {
                if (TRANS_B) {
                    bF[ni] = *(const v2f*)(&sB[cur][(waveN + ni * 16 + ln) * LDSA + kk]);
                } else {
                    v2f b;
                    b.x = sB[cur][(kk + 0) * LDSB_NN + (waveN + ni * 16 + ln)];
                    b.y = sB[cur][(kk + 1) * LDSB_NN + (waveN + ni * 16 + ln)];
                    bF[ni] = b;
                }
            }
#pragma unroll
            for (int mi = 0; mi < 4; ++mi)
#pragma unroll
                for (int ni = 0; ni < 2; ++ni)
                    acc[mi][ni] = __builtin_amdgcn_wmma_f32_16x16x4_f32(
                        false, aF[mi], false, bF[ni],
                        (short)0, acc[mi][ni], false, false);
        }

        wait_async_all();    // next tile landed (no-op on final iteration)
        __syncthreads();     // all waves done reading `cur`, next buffer visible
    }

    // ---- epilogue + store (C/D layout: VGPR r -> M = r + 8*khalf, N = ln) ----
    const int rowBase = bm + waveM;
    const int colBase = bn + waveN;
#pragma unroll
    for (int mi = 0; mi < 4; ++mi) {
#pragma unroll
        for (int ni = 0; ni < 2; ++ni) {
#pragma unroll
            for (int r = 0; r < 8; ++r) {
                const int row = rowBase + mi * 16 + r + khalf * 8;
                const int col = colBase + ni * 16 + ln;
                float v = acc[mi][ni][r];
                if (EPI == 1) v = alpha * v + beta * aux[(size_t)row * Ndim + col];
                else if (EPI == 2) v = v + aux[col];
                D[(size_t)row * Ndim + col] = v;
            }
        }
    }
}

// ---------------------------------------------------------------------------
// Elementwise / reduction helpers
// ---------------------------------------------------------------------------
__global__ void build_M_kernel(const float* __restrict__ w, float* __restrict__ Mo, int n)
{
    const int j = blockIdx.x * blockDim.x + threadIdx.x;
    const int i = blockIdx.y;
    if (j < n) {
        const float a = w[(size_t)i * n + j] - w[(size_t)j * n + i]; // A = W - W^T
        Mo[(size_t)i * n + j] = (i == j ? 1.0f : 0.0f) - a;          // M = I - A
    }
}

__global__ void reduce_sq_kernel(const float* __restrict__ p, float* __restrict__ partial, int n)
{
    __shared__ float s[256];
    const int tid = threadIdx.x;
    float acc = 0.f;
    for (int i = blockIdx.x * 256 + tid; i < n; i += gridDim.x * 256) {
        const float v = p[i];
        acc += v * v;
    }
    s[tid] = acc;
    __syncthreads();
    for (int o = 128; o > 0; o >>= 1) {
        if (tid < o) s[tid] += s[tid + o];
        __syncthreads();
    }
    if (tid == 0) partial[blockIdx.x] = s[0];
}

__global__ void finalize_scale_kernel(const float* __restrict__ partial,
                                      float* __restrict__ scale, int nblocks)
{
    __shared__ float s[256];
    const int tid = threadIdx.x;
    float acc = 0.f;
    for (int i = tid; i < nblocks; i += 256) acc += partial[i];
    s[tid] = acc;
    __syncthreads();
    for (int o = 128; o > 0; o >>= 1) {
        if (tid < o) s[tid] += s[tid + o];
        __syncthreads();
    }
    if (tid == 0) scale[0] = 1.0f / s[0];   // c = 1 / ||M||_F^2
}

__global__ void init_X_kernel(const float* __restrict__ Mo, const float* __restrict__ scale,
                              float* __restrict__ X, int n)
{
    const int j = blockIdx.x * blockDim.x + threadIdx.x;
    const int i = blockIdx.y;
    const float c = scale[0];
    if (j < n) X[(size_t)i * n + j] = c * Mo[(size_t)j * n + i];  // X0 = M^T / ||M||_F^2
}

// ---------------------------------------------------------------------------
// Orchestration
// ---------------------------------------------------------------------------
extern "C" void kernel_launch(void* const* d_in, const int* in_sizes, int n_in,
                              void* d_out, int out_size, void* d_ws, size_t ws_size,
                              hipStream_t stream)
{
    (void)in_sizes; (void)n_in; (void)out_size; (void)ws_size;

    const float* x    = (const float*)d_in[0];   // (4,2048,2048)
    const float* w    = (const float*)d_in[1];   // (2048,2048)
    const float* bias = (const float*)d_in[2];   // (2048,)
    float* out = (float*)d_out;                  // (4,2048,2048)

    const int n = DIM;
    const size_t nn = (size_t)n * n;

    float* ws    = (float*)d_ws;
    float* Mo    = ws;            // M = I - A
    float* X     = ws + 1 * nn;   // Newton iterate (== M^{-1} at convergence)
    float* Xn    = ws + 2 * nn;
    float* T     = ws + 3 * nn;   // T = M * X
    float* Wm    = ws + 4 * nn;   // W* = X * M^T
    float* red   = ws + 5 * nn;   // 1024 partial sums
    float* scale = red + 1024;    // 1 scalar

    const dim3 b256(256);

    build_M_kernel       <<<dim3(n / 256, n), b256, 0, stream>>>(w, Mo, n);
    reduce_sq_kernel     <<<dim3(1024),       b256, 0, stream>>>(Mo, red, (int)nn);
    finalize_scale_kernel<<<dim3(1),          b256, 0, stream>>>(red, scale, 1024);
    init_X_kernel        <<<dim3(n / 256, n), b256, 0, stream>>>(Mo, scale, X, n);

    const dim3 gS(n / TM, n / TN);   // 16 x 16 blocks for 2048^2 GEMMs
    for (int it = 0; it < NEWTON_ITERS; ++it) {
        // T  = M * X
        gemm_f32_wmma<false, 0><<<gS, b256, 0, stream>>>(Mo, X, T, nullptr, 1.f, 0.f, n, n, n);
        // Xn = 2*X - X*T  =  (-1)*X*T + 2*C  with C = X
        gemm_f32_wmma<false, 1><<<gS, b256, 0, stream>>>(X, T, Xn, X, -1.f, 2.f, n, n, n);
        float* tmp = X; X = Xn; Xn = tmp;
    }

    // W* = X * M^T     (since I + A == (I - A)^T)
    gemm_f32_wmma<true, 0><<<gS, b256, 0, stream>>>(X, Mo, Wm, nullptr, 1.f, 0.f, n, n, n);

    // out = x * W*^T + bias
    const dim3 gO(BATCH_ROWS / TM, n / TN);  // 64 x 16
    gemm_f32_wmma<true, 2><<<gO, b256, 0, stream>>>(x, Wm, out, bias, 1.f, 0.f,
                                                    BATCH_ROWS, n, n);
}